// GraphSAGE_26852135535160
// MI455X (gfx1250) — compile-verified
//
#include <hip/hip_runtime.h>

#define N_NODES 50000
#define N_EDGES 800000
#define DFEAT   128
#define LDSW    132   // padded LDS row stride (floats) to avoid bank conflicts

typedef __attribute__((ext_vector_type(2))) float v2f;
typedef __attribute__((ext_vector_type(8))) float v8f;

// ---------------- degree kernels ----------------
__global__ __launch_bounds__(256) void deg_count_kernel(const int* __restrict__ ei,
                                                        float* __restrict__ deg) {
  int e = blockIdx.x * 256 + threadIdx.x;
  if (e < N_EDGES) {
    int dst = ei[N_EDGES + e];
    unsafeAtomicAdd(&deg[dst], 1.0f);
  }
}

__global__ __launch_bounds__(256) void deg_inv_kernel(float* __restrict__ deg) {
  int i = blockIdx.x * 256 + threadIdx.x;
  if (i < N_NODES) deg[i] = 1.0f / fmaxf(deg[i], 1.0f);
}

// ---------------- edge scatter (mean-agg numerator) ----------------
// One wave (32 lanes) per edge; lane reads float4 of source row, 4 f32 atomics to dst row.
// Grid is exactly E/8 blocks of 256 threads (E divisible by 8) -> no bounds check,
// EXEC stays all-ones.
__global__ __launch_bounds__(256) void scatter_kernel(const int* __restrict__ ei,
                                                      const float* __restrict__ x,
                                                      float* __restrict__ agg) {
  unsigned gid  = blockIdx.x * 256u + threadIdx.x;
  unsigned e    = gid >> 5;
  unsigned lane = gid & 31u;
  int src = ei[e];
  int dst = ei[N_EDGES + e];
  const float4 v = ((const float4*)(x + (size_t)src * DFEAT))[lane];
  float* p = agg + (size_t)dst * DFEAT + (size_t)lane * 4;
  unsafeAtomicAdd(p + 0, v.x);
  unsafeAtomicAdd(p + 1, v.y);
  unsafeAtomicAdd(p + 2, v.z);
  unsafeAtomicAdd(p + 3, v.w);
}

// ---------------- fused SAGE layer GEMM ----------------
// out[16 x 128 tile] = relu( (agg*deg_inv) @ Wl^T + bl + x @ Wr^T )
// Block: 256 threads = 8 waves; wave w computes cols [16w, 16w+16) of the 16-row tile.
// f32 WMMA 16x16x4: A lanes 0-15 -> K={0,1}, lanes 16-31 -> K={2,3}, M=lane%16.
__global__ __launch_bounds__(256) void sage_gemm_kernel(
    const float* __restrict__ xin, const float* __restrict__ agg,
    const float* __restrict__ deg_inv,
    const float* __restrict__ Wl, const float* __restrict__ bl,
    const float* __restrict__ Wr, float* __restrict__ xout) {
  __shared__ float s_a[16 * LDSW];
  __shared__ float s_x[16 * LDSW];

  int tid = threadIdx.x;
  // ---- stage 16x128 tiles of agg (pre-scaled) and x into LDS ----
  int row = tid >> 4;            // 0..15
  int c0  = (tid & 15) * 8;      // 8 floats per thread
  size_t node = (size_t)blockIdx.x * 16 + row;
  float dinv = deg_inv[node];
  const float4* pa = (const float4*)(agg + node * DFEAT + c0);
  const float4* px = (const float4*)(xin + node * DFEAT + c0);
  float4 a0 = pa[0], a1 = pa[1];
  float4 x0 = px[0], x1 = px[1];
  float* sa = &s_a[row * LDSW + c0];
  float* sx = &s_x[row * LDSW + c0];
  sa[0] = a0.x * dinv; sa[1] = a0.y * dinv; sa[2] = a0.z * dinv; sa[3] = a0.w * dinv;
  sa[4] = a1.x * dinv; sa[5] = a1.y * dinv; sa[6] = a1.z * dinv; sa[7] = a1.w * dinv;
  sx[0] = x0.x; sx[1] = x0.y; sx[2] = x0.z; sx[3] = x0.w;
  sx[4] = x1.x; sx[5] = x1.y; sx[6] = x1.z; sx[7] = x1.w;
  __syncthreads();

  // ---- per-wave 16x16 WMMA tile ----
  int lane = tid & 31;
  int w    = tid >> 5;           // wave id 0..7
  int half = lane >> 4;          // 0: K pair {0,1}; 1: K pair {2,3}
  int m    = lane & 15;          // A row / B column within slab
  int j0   = w * 16;             // output-column slab

  v8f acc = {0.f, 0.f, 0.f, 0.f, 0.f, 0.f, 0.f, 0.f};
  const float* wlp  = Wl + (size_t)(j0 + m) * DFEAT + 2 * half;  // B[k][j] = Wl[j][k]
  const float* wrp  = Wr + (size_t)(j0 + m) * DFEAT + 2 * half;
  const float* arow = &s_a[m * LDSW + 2 * half];
  const float* xrow = &s_x[m * LDSW + 2 * half];

#pragma unroll
  for (int k = 0; k < DFEAT; k += 4) {
    v2f a, b;
    a.x = arow[k]; a.y = arow[k + 1];
    b.x = wlp[k];  b.y = wlp[k + 1];
    acc = __builtin_amdgcn_wmma_f32_16x16x4_f32(false, a, false, b, (short)0, acc,
                                                false, false);
  }
#pragma unroll
  for (int k = 0; k < DFEAT; k += 4) {
    v2f a, b;
    a.x = xrow[k]; a.y = xrow[k + 1];
    b.x = wrp[k];  b.y = wrp[k + 1];
    acc = __builtin_amdgcn_wmma_f32_16x16x4_f32(false, a, false, b, (short)0, acc,
                                                false, false);
  }

  // ---- bias + relu + store (C/D: VGPR v -> rows v and v+8; col = j0 + m) ----
  float bias = bl[j0 + m];
  size_t rbase = (size_t)blockIdx.x * 16 + (size_t)half * 8;
#pragma unroll
  for (int v = 0; v < 8; v++) {
    float val = acc[v] + bias;
    xout[(rbase + v) * DFEAT + j0 + m] = fmaxf(val, 0.0f);
  }
}

// ---------------- launcher ----------------
extern "C" void kernel_launch(void* const* d_in, const int* in_sizes, int n_in,
                              void* d_out, int out_size, void* d_ws, size_t ws_size,
                              hipStream_t stream) {
  (void)in_sizes; (void)n_in; (void)out_size; (void)ws_size;
  const float* xfeat = (const float*)d_in[0];
  const int*   ei    = (const int*)d_in[1];
  const float* Wl[3] = {(const float*)d_in[2], (const float*)d_in[5], (const float*)d_in[8]};
  const float* bl[3] = {(const float*)d_in[3], (const float*)d_in[6], (const float*)d_in[9]};
  const float* Wr[3] = {(const float*)d_in[4], (const float*)d_in[7], (const float*)d_in[10]};
  float* out = (float*)d_out;

  // workspace layout (d_ws is 256B-aligned from hipMalloc)
  const size_t NPAD = 50048;                       // N rounded up, keeps 256B alignment
  float* deg = (float*)d_ws;                       // N floats (becomes deg_inv in place)
  float* agg = deg + NPAD;                         // N*128 floats
  float* xA  = agg + (size_t)N_NODES * DFEAT;      // N*128 floats
  float* xB  = xA  + (size_t)N_NODES * DFEAT;      // N*128 floats

  hipMemsetAsync(deg, 0, (size_t)N_NODES * sizeof(float), stream);
  deg_count_kernel<<<(N_EDGES + 255) / 256, 256, 0, stream>>>(ei, deg);
  deg_inv_kernel<<<(N_NODES + 255) / 256, 256, 0, stream>>>(deg);

  const float* xin = xfeat;
  float* outs[3] = {xA, xB, out};
  for (int l = 0; l < 3; ++l) {
    hipMemsetAsync(agg, 0, (size_t)N_NODES * DFEAT * sizeof(float), stream);
    scatter_kernel<<<N_EDGES / 8, 256, 0, stream>>>(ei, xin, agg);
    sage_gemm_kernel<<<N_NODES / 16, 256, 0, stream>>>(xin, agg, deg,
                                                       Wl[l], bl[l], Wr[l], outs[l]);
    xin = outs[l];
  }
}